// ADMLayer_19284403159340
// MI455X (gfx1250) — compile-verified
//
#include <hip/hip_runtime.h>
#include <math.h>

#define E_    2
#define B_    75
#define C_    64
#define HW_   196
#define WAY_  5
#define SHOT_ 5
#define S_    (WAY_ * SHOT_)   // 25
#define SN_   (SHOT_ * HW_)    // 980
#define LDA_  66               // stride for 64x64 matrices in LDS
#define QS_   208              // padded pixel-column stride (13 N/M tiles)
#define PS_   210              // product-buffer row stride

typedef __attribute__((ext_vector_type(2))) float v2f;
typedef __attribute__((ext_vector_type(8))) float v8f;

__device__ __forceinline__ v8f wmma_f32(v2f a, v2f b, v8f c) {
  // D(16x16,f32) = A(16x4,f32) * B(4x16,f32) + C
  return __builtin_amdgcn_wmma_f32_16x16x4_f32(false, a, false, b, (short)0, c,
                                               false, false);
}

// Async global->LDS copy of one 16-byte chunk. LDS address is the low 32 bits
// of the generic pointer (generic LDS pointers carry the LDS offset in [31:0]).
__device__ __forceinline__ void async_b128(const float* g, float* l) {
  unsigned lp = (unsigned)(unsigned long long)l;
  asm volatile("global_load_async_to_lds_b128 %0, %1, off"
               :: "v"(lp), "v"(g) : "memory");
}
__device__ __forceinline__ void async_wait0() {
  asm volatile("s_wait_asynccnt 0x0" ::: "memory");
}

// Flat contiguous copy (nfloats multiple of 4; bases 16B aligned).
__device__ __forceinline__ void async_copy_flat(const float* g, float* l,
                                                int nfloats, int tid) {
  for (int c4 = tid * 4; c4 < nfloats; c4 += 256 * 4) async_b128(g + c4, l + c4);
  async_wait0();
}

// Strided 2D copy: nrows x ncols floats (ncols multiple of 4).
__device__ __forceinline__ void async_copy_2d(const float* g, int gstride,
                                              float* l, int lstride, int nrows,
                                              int ncols, int tid) {
  const int cq = ncols >> 2;
  for (int idx = tid; idx < nrows * cq; idx += 256) {
    int r = idx / cq, c4 = (idx - r * cq) << 2;
    async_b128(g + r * gstride + c4, l + r * lstride + c4);
  }
  async_wait0();
}

// ---------------------------------------------------------------------------
// Kernel 1: per (e,way) support stats: mean, cov (ridge), inverse, logdet.
// X kept in global layout [ch][pix]; gram = X * X^T via f32 WMMA (NT pattern).
// ---------------------------------------------------------------------------
__global__ void __launch_bounds__(256)
support_prep_kernel(const float* __restrict__ sf, float* __restrict__ s_mean,
                    float* __restrict__ s_covinv, float* __restrict__ ld2) {
  const int e = blockIdx.x / WAY_, w = blockIdx.x % WAY_;
  __shared__ __align__(16) float X[C_ * HW_];  // [ch][pix], stride 196
  __shared__ float A[C_ * LDA_];               // gram -> cov -> GJ-eliminated
  __shared__ float Iv[C_ * LDA_];              // identity -> inverse
  __shared__ float rowsum[C_];
  __shared__ float fac[C_];
  __shared__ float ldacc;
  const int tid = threadIdx.x;
  const int lane = tid & 31, wv = tid >> 5;
  const int lo = lane & 15, hi = lane >> 4;
  if (tid < C_) rowsum[tid] = 0.f;
  if (tid == 0) ldacc = 0.f;
  // wave wv owns gram tiles wv and wv+8 (of 16); accumulators live in regs
  const int ti0 = (wv & 3) * 16, tj0 = (wv >> 2) * 16;
  const int ti1 = ((wv + 8) & 3) * 16, tj1 = ((wv + 8) >> 2) * 16;
  v8f acc0 = {}, acc1 = {};
  for (int shot = 0; shot < SHOT_; ++shot) {
    __syncthreads();  // readers of previous chunk done before overwrite
    const long base = (long)((e * S_ + w * SHOT_ + shot) * C_) * HW_;
    async_copy_flat(sf + base, X, C_ * HW_, tid);
    __syncthreads();
    if (tid < C_) {  // per-channel sums (row sums in this layout)
      float s = 0.f;
      for (int p = 0; p < HW_; ++p) s += X[tid * HW_ + p];
      rowsum[tid] += s;
    }
    for (int k0 = 0; k0 < HW_; k0 += 4) {  // 49 K-steps per chunk
      int kr = k0 + 2 * hi;
      v2f a0 = *(const v2f*)&X[(ti0 + lo) * HW_ + kr];
      v2f b0 = *(const v2f*)&X[(tj0 + lo) * HW_ + kr];
      acc0 = wmma_f32(a0, b0, acc0);
      v2f a1 = *(const v2f*)&X[(ti1 + lo) * HW_ + kr];
      v2f b1 = *(const v2f*)&X[(tj1 + lo) * HW_ + kr];
      acc1 = wmma_f32(a1, b1, acc1);
    }
  }
  __syncthreads();
  for (int r = 0; r < 8; ++r) {
    A[(ti0 + r + 8 * hi) * LDA_ + tj0 + lo] = acc0[r];
    A[(ti1 + r + 8 * hi) * LDA_ + tj1 + lo] = acc1[r];
  }
  if (tid < C_) rowsum[tid] *= (1.f / (float)SN_);  // mean
  __syncthreads();
  for (int idx = tid; idx < C_ * C_; idx += 256) {  // cov + ridge, init I
    int i = idx >> 6, j = idx & 63;
    float c = (A[i * LDA_ + j] - (float)SN_ * rowsum[i] * rowsum[j]) *
              (1.f / (float)(SN_ - 1));
    if (i == j) c += 0.01f;
    A[i * LDA_ + j] = c;
    Iv[i * LDA_ + j] = (i == j) ? 1.f : 0.f;
  }
  __syncthreads();
  // Gauss-Jordan (SPD + ridge -> no pivoting); logdet = sum log pivots
  for (int k = 0; k < C_; ++k) {
    float piv = A[k * LDA_ + k];
    float invp = 1.f / piv;
    __syncthreads();
    if (tid == 0) ldacc += logf(piv);
    if (tid < C_) {
      A[k * LDA_ + tid] *= invp;
      Iv[k * LDA_ + tid] *= invp;
      if (tid != k) fac[tid] = A[tid * LDA_ + k];
    }
    __syncthreads();
    for (int idx = tid; idx < C_ * C_; idx += 256) {
      int i = idx >> 6, j = idx & 63;
      if (i != k) {
        float f = fac[i];
        A[i * LDA_ + j]  -= f * A[k * LDA_ + j];
        Iv[i * LDA_ + j] -= f * Iv[k * LDA_ + j];
      }
    }
    __syncthreads();
  }
  const int ew = e * WAY_ + w;
  for (int idx = tid; idx < C_ * C_; idx += 256) {
    int i = idx >> 6, j = idx & 63;
    s_covinv[(long)ew * 4096 + idx] = Iv[i * LDA_ + j];
  }
  if (tid < C_) s_mean[ew * C_ + tid] = rowsum[tid];
  if (tid == 0) ld2[ew] = ldacc;
}

// ---------------------------------------------------------------------------
// Kernel 2: per (e,b) query stats: mean, cov (ridge) -> global, logdet.
// ---------------------------------------------------------------------------
__global__ void __launch_bounds__(256)
query_prep_kernel(const float* __restrict__ qf, float* __restrict__ q_mean,
                  float* __restrict__ q_cov, float* __restrict__ ld1) {
  const int eb = blockIdx.x;
  __shared__ __align__(16) float X[C_ * HW_];  // [ch][pix]
  __shared__ float A[C_ * LDA_];
  __shared__ float rowsum[C_];
  __shared__ float fac[C_];
  __shared__ float ldacc;
  const int tid = threadIdx.x;
  const int lane = tid & 31, wv = tid >> 5;
  const int lo = lane & 15, hi = lane >> 4;
  if (tid == 0) ldacc = 0.f;
  async_copy_flat(qf + (long)eb * C_ * HW_, X, C_ * HW_, tid);
  __syncthreads();
  if (tid < C_) {
    float s = 0.f;
    for (int p = 0; p < HW_; ++p) s += X[tid * HW_ + p];
    rowsum[tid] = s * (1.f / (float)HW_);
  }
  const int ti0 = (wv & 3) * 16, tj0 = (wv >> 2) * 16;
  const int ti1 = ((wv + 8) & 3) * 16, tj1 = ((wv + 8) >> 2) * 16;
  v8f acc0 = {}, acc1 = {};
  for (int k0 = 0; k0 < HW_; k0 += 4) {
    int kr = k0 + 2 * hi;
    v2f a0 = *(const v2f*)&X[(ti0 + lo) * HW_ + kr];
    v2f b0 = *(const v2f*)&X[(tj0 + lo) * HW_ + kr];
    acc0 = wmma_f32(a0, b0, acc0);
    v2f a1 = *(const v2f*)&X[(ti1 + lo) * HW_ + kr];
    v2f b1 = *(const v2f*)&X[(tj1 + lo) * HW_ + kr];
    acc1 = wmma_f32(a1, b1, acc1);
  }
  __syncthreads();
  for (int r = 0; r < 8; ++r) {
    A[(ti0 + r + 8 * hi) * LDA_ + tj0 + lo] = acc0[r];
    A[(ti1 + r + 8 * hi) * LDA_ + tj1 + lo] = acc1[r];
  }
  __syncthreads();
  for (int idx = tid; idx < C_ * C_; idx += 256) {
    int i = idx >> 6, j = idx & 63;
    float c = (A[i * LDA_ + j] - (float)HW_ * rowsum[i] * rowsum[j]) *
              (1.f / (float)(HW_ - 1));
    if (i == j) c += 0.01f;
    A[i * LDA_ + j] = c;
    q_cov[(long)eb * 4096 + idx] = c;
  }
  __syncthreads();
  // forward Gaussian elimination for logdet only
  for (int k = 0; k < C_; ++k) {
    float piv = A[k * LDA_ + k];
    if (tid == 0) ldacc += logf(piv);
    if (tid < C_ && tid > k) fac[tid] = A[tid * LDA_ + k] / piv;
    __syncthreads();
    for (int idx = tid; idx < C_ * C_; idx += 256) {
      int i = idx >> 6, j = idx & 63;
      if (i > k) A[i * LDA_ + j] -= fac[i] * A[k * LDA_ + j];
    }
    __syncthreads();
  }
  if (tid < C_) q_mean[eb * C_ + tid] = rowsum[tid];
  if (tid == 0) ld1[eb] = ldacc;
}

// ---------------------------------------------------------------------------
// Kernel 3: per (e,b,way) KL distance -> adm[.., way]
// ---------------------------------------------------------------------------
__global__ void __launch_bounds__(256)
kl_kernel(const float* __restrict__ q_mean, const float* __restrict__ q_cov,
          const float* __restrict__ ld1, const float* __restrict__ s_mean,
          const float* __restrict__ s_covinv, const float* __restrict__ ld2,
          float* __restrict__ adm) {
  const int w = blockIdx.x % WAY_;
  const int eb = blockIdx.x / WAY_;
  const int e = eb / B_;
  const int ew = e * WAY_ + w;
  const float* qc = q_cov + (long)eb * 4096;
  const float* mi = s_covinv + (long)ew * 4096;
  __shared__ float d[C_];
  __shared__ float red[256];
  const int tid = threadIdx.x;
  if (tid < C_) d[tid] = s_mean[ew * C_ + tid] - q_mean[eb * C_ + tid];
  __syncthreads();
  float part = 0.f;
  for (int idx = tid; idx < 4096; idx += 256) {  // tr(q_cov * inv)
    int i = idx >> 6, j = idx & 63;
    part += qc[idx] * mi[j * 64 + i];
  }
  if (tid < C_) {  // maha
    float tj = 0.f;
    for (int i = 0; i < C_; ++i) tj += d[i] * mi[i * 64 + tid];
    part += tj * d[tid];
  }
  red[tid] = part;
  __syncthreads();
  for (int off = 128; off > 0; off >>= 1) {
    if (tid < off) red[tid] += red[tid + off];
    __syncthreads();
  }
  if (tid == 0) {
    float kl = 0.5f * (red[0] + ld2[ew] - ld1[eb] - (float)C_);
    adm[eb * (2 * WAY_) + w] = -kl;
  }
}

// ---------------------------------------------------------------------------
// Kernel 4: per (e,b,way) cosine inner products via f32 WMMA + per-row top-3.
// Panels held in global layout [ch][pix], async-staged; one shot per chunk.
// LDS: Q 64x208 + S 64x208 + P 208x210  (~282 KB of the 320 KB WGP budget)
// ---------------------------------------------------------------------------
__global__ void __launch_bounds__(256)
topk_kernel(const float* __restrict__ qf, const float* __restrict__ sf,
            float* __restrict__ adm) {
  const int w = blockIdx.x % WAY_;
  const int eb = blockIdx.x / WAY_;
  const int e = eb / B_;
  __shared__ __align__(16) float Q[C_ * QS_];
  __shared__ __align__(16) float Sh[C_ * QS_];
  __shared__ float P[QS_ * PS_];
  __shared__ float red[256];
  const int tid = threadIdx.x;
  const int lane = tid & 31, wv = tid >> 5;
  const int lo = lane & 15, hi = lane >> 4;
  // zero Q pad columns (196..207), then async-stage query panel
  for (int idx = tid; idx < C_ * (QS_ - HW_); idx += 256) {
    int c = idx / (QS_ - HW_), j = idx - c * (QS_ - HW_);
    Q[c * QS_ + HW_ + j] = 0.f;
  }
  async_copy_2d(qf + (long)eb * C_ * HW_, HW_, Q, QS_, C_, HW_, tid);
  __syncthreads();
  if (tid < QS_) {  // per-descriptor (column) L2 normalize; pads stay zero
    float ss = 0.f;
    for (int c = 0; c < C_; ++c) { float v = Q[c * QS_ + tid]; ss += v * v; }
    float r = 1.f / fmaxf(sqrtf(ss), 1e-12f);
    for (int c = 0; c < C_; ++c) Q[c * QS_ + tid] *= r;
  }
  float t0 = -1e30f, t1 = -1e30f, t2 = -1e30f;  // running top-3 per row
  const int sbase = e * S_ + w * SHOT_;
  for (int shot = 0; shot < SHOT_; ++shot) {  // one 196-col chunk per shot
    __syncthreads();
    for (int idx = tid; idx < C_ * (QS_ - HW_); idx += 256) {
      int c = idx / (QS_ - HW_), j = idx - c * (QS_ - HW_);
      Sh[c * QS_ + HW_ + j] = 0.f;
    }
    async_copy_2d(sf + (long)(sbase + shot) * C_ * HW_, HW_, Sh, QS_, C_, HW_,
                  tid);
    __syncthreads();
    if (tid < QS_) {
      float ss = 0.f;
      for (int c = 0; c < C_; ++c) { float v = Sh[c * QS_ + tid]; ss += v * v; }
      float r = 1.f / fmaxf(sqrtf(ss), 1e-12f);
      for (int c = 0; c < C_; ++c) Sh[c * QS_ + tid] *= r;
    }
    __syncthreads();
    // 13 M-tiles x 13 N-tiles = 169 jobs; uniform within wave -> EXEC full
    for (int job = wv; job < 169; job += 8) {
      int mt = job % 13, nt = job / 13;
      v8f acc = {};
      for (int k0 = 0; k0 < C_; k0 += 4) {
        int kr = k0 + 2 * hi;
        v2f a, b;
        a.x = Q[kr * QS_ + mt * 16 + lo];
        a.y = Q[(kr + 1) * QS_ + mt * 16 + lo];
        b.x = Sh[kr * QS_ + nt * 16 + lo];
        b.y = Sh[(kr + 1) * QS_ + nt * 16 + lo];
        acc = wmma_f32(a, b, acc);
      }
      float* prow = &P[(mt * 16 + 8 * hi) * PS_ + nt * 16 + lo];
      for (int r = 0; r < 8; ++r) prow[r * PS_] = acc[r];
    }
    __syncthreads();
    if (tid < HW_) {  // thread tid owns descriptor row tid
      const float* pr = &P[tid * PS_];
      for (int j = 0; j < HW_; ++j) {
        float v = pr[j];
        if (v > t2) {
          if (v > t1) {
            t2 = t1;
            if (v > t0) { t1 = t0; t0 = v; } else { t1 = v; }
          } else {
            t2 = v;
          }
        }
      }
    }
  }
  __syncthreads();
  red[tid] = (tid < HW_) ? (t0 + t1 + t2) : 0.f;
  __syncthreads();
  for (int off = 128; off > 0; off >>= 1) {
    if (tid < off) red[tid] += red[tid + off];
    __syncthreads();
  }
  if (tid == 0) adm[eb * (2 * WAY_) + WAY_ + w] = red[0];
}

// ---------------------------------------------------------------------------
// Kernel 5: per episode e: BatchNorm over b, then dilated conv1d fuse.
// ---------------------------------------------------------------------------
__global__ void __launch_bounds__(256)
fuse_kernel(const float* __restrict__ adm, const float* __restrict__ gamma,
            const float* __restrict__ beta, const float* __restrict__ cw,
            float* __restrict__ out) {
  const int e = blockIdx.x;
  __shared__ float a[B_ * 2 * WAY_];
  __shared__ float scale[2 * WAY_], shift[2 * WAY_];
  const int tid = threadIdx.x;
  for (int idx = tid; idx < B_ * 2 * WAY_; idx += 256)
    a[idx] = adm[e * B_ * 2 * WAY_ + idx];
  __syncthreads();
  if (tid < 2 * WAY_) {
    float s = 0.f, ss = 0.f;
    for (int b = 0; b < B_; ++b) {
      float v = a[b * 2 * WAY_ + tid];
      s += v; ss += v * v;
    }
    float m = s * (1.f / (float)B_);
    float var = ss * (1.f / (float)B_) - m * m;  // biased, like torch BN
    float sc = rsqrtf(var + 1e-5f) * gamma[tid];
    scale[tid] = sc;
    shift[tid] = beta[tid] - m * sc;
  }
  __syncthreads();
  float w0 = cw[0], w1 = cw[1];
  for (int idx = tid; idx < B_ * WAY_; idx += 256) {
    int b = idx / WAY_, k = idx - b * WAY_;
    float x0 = a[b * 2 * WAY_ + k] * scale[k] + shift[k];
    float x1 = a[b * 2 * WAY_ + k + WAY_] * scale[k + WAY_] + shift[k + WAY_];
    out[(e * B_ + b) * WAY_ + k] = w0 * x0 + w1 * x1;
  }
}

// ---------------------------------------------------------------------------
extern "C" void kernel_launch(void* const* d_in, const int* in_sizes, int n_in,
                              void* d_out, int out_size, void* d_ws,
                              size_t ws_size, hipStream_t stream) {
  (void)in_sizes; (void)n_in; (void)out_size; (void)ws_size;
  const float* qf    = (const float*)d_in[0];  // [2,75,64,14,14]
  const float* sf    = (const float*)d_in[1];  // [2,25,64,14,14]
  const float* gamma = (const float*)d_in[2];  // [10]
  const float* beta  = (const float*)d_in[3];  // [10]
  const float* cw    = (const float*)d_in[4];  // [2]
  float* out = (float*)d_out;                  // [2,75,5]

  float* ws       = (float*)d_ws;
  float* s_mean   = ws;                                  // 2*5*64
  float* s_covinv = s_mean + E_ * WAY_ * C_;             // 2*5*4096
  float* ld2      = s_covinv + E_ * WAY_ * C_ * C_;      // 10
  float* q_mean   = ld2 + E_ * WAY_;                     // 2*75*64
  float* q_cov    = q_mean + E_ * B_ * C_;               // 2*75*4096
  float* ld1      = q_cov + (size_t)E_ * B_ * C_ * C_;   // 150
  float* adm      = ld1 + E_ * B_;                       // 2*75*10

  support_prep_kernel<<<E_ * WAY_, 256, 0, stream>>>(sf, s_mean, s_covinv, ld2);
  query_prep_kernel<<<E_ * B_, 256, 0, stream>>>(qf, q_mean, q_cov, ld1);
  kl_kernel<<<E_ * B_ * WAY_, 256, 0, stream>>>(q_mean, q_cov, ld1, s_mean,
                                                s_covinv, ld2, adm);
  topk_kernel<<<E_ * B_ * WAY_, 256, 0, stream>>>(qf, sf, adm);
  fuse_kernel<<<E_, 256, 0, stream>>>(adm, gamma, beta, cw, out);
}